// GNN_24678882082891
// MI455X (gfx1250) — compile-verified
//
#include <hip/hip_runtime.h>
#include <cstdint>

typedef __attribute__((ext_vector_type(2))) float v2f;
typedef __attribute__((ext_vector_type(8))) float v8f;

// ---------------------------------------------------------------------------
// f32 WMMA GEMM: C[M,N] = op(A)[M,K] @ B[K,N], row-major. M%16==0, K%4==0,
// N % (16*CT) == 0. One wave computes a 16 x (16*CT) C strip using CT
// independent V_WMMA_F32_16X16X4_F32 accumulator chains: the A fragment is
// loaded once per k-step and reused CT times (cuts A traffic by CT and gives
// the XDL pipe CT-way ILP instead of one serial RAW chain).
// RELU_A applies relu to A elements on load (fuses the reference's relu()).
// ---------------------------------------------------------------------------
template <bool RELU_A, int CT>
__global__ void gat_gemm_f32_wmma(const float* __restrict__ A,
                                  const float* __restrict__ B,
                                  float* __restrict__ C,
                                  int M, int N, int K) {
  const int lane = threadIdx.x & 31;
  const int wave = blockIdx.x * (blockDim.x >> 5) + (threadIdx.x >> 5);
  const int colGroups = N / (16 * CT);
  const int totalTiles = (M >> 4) * colGroups;
  if (wave >= totalTiles) return;  // wave-uniform: EXEC stays all-1s for WMMA

  const int tr = wave / colGroups;
  const int tg = wave - tr * colGroups;
  const int row0 = tr << 4;
  const int col0 = tg * (16 * CT);
  const int half = lane >> 4;   // selects K-pair (0,1) vs (2,3)
  const int l16  = lane & 15;   // A: row M ; B/C: col N

  // A: lane holds A[row0+l16][k0+2*half .. +1] in a.x/a.y  (8B aligned)
  const float2* A2 =
      reinterpret_cast<const float2*>(A + (size_t)(row0 + l16) * K + 2 * half);
  // B: lane holds B[k0+2*half][col0+16*j+l16] in b.x, next K-row in b.y
  const float* Bcol = B + (size_t)(2 * half) * N + col0 + l16;

  v8f acc[CT];
#pragma unroll
  for (int j = 0; j < CT; ++j) acc[j] = (v8f){};

  for (int k0 = 0; k0 < K; k0 += 4) {
    float2 av = A2[k0 >> 1];
    if (RELU_A) { av.x = fmaxf(av.x, 0.f); av.y = fmaxf(av.y, 0.f); }
    v2f a;
    a.x = av.x; a.y = av.y;
    const float* b0 = Bcol + (size_t)k0 * N;
    const float* b1 = Bcol + (size_t)(k0 + 1) * N;
#pragma unroll
    for (int j = 0; j < CT; ++j) {
      v2f b;
      b.x = b0[16 * j];
      b.y = b1[16 * j];
      acc[j] = __builtin_amdgcn_wmma_f32_16x16x4_f32(false, a, false, b,
                                                     (short)0, acc[j],
                                                     false, false);
    }
  }

  // C: lanes 0-15 -> M=r, lanes 16-31 -> M=8+r ; N = col0+16*j+l16
  float* Crow = C + (size_t)(row0 + half * 8) * N + col0 + l16;
#pragma unroll
  for (int j = 0; j < CT; ++j)
#pragma unroll
    for (int r = 0; r < 8; ++r) Crow[(size_t)r * N + 16 * j] = acc[j][r];
}

// ---------------------------------------------------------------------------
// e_src[i] = Wx[i,:] . a_src ; e_dst[i] = Wx[i,:] . a_dst. One wave per node.
// ---------------------------------------------------------------------------
__global__ void gat_gemv2(const float* __restrict__ Wx,
                          const float* __restrict__ a_s,
                          const float* __restrict__ a_d,
                          float* __restrict__ es, float* __restrict__ ed,
                          int Nn, int H) {
  const int lane = threadIdx.x & 31;
  const int node = blockIdx.x * (blockDim.x >> 5) + (threadIdx.x >> 5);
  if (node >= Nn) return;
  const float* row = Wx + (size_t)node * H;
  float ss = 0.f, sd = 0.f;
  for (int f = lane; f < H; f += 32) {
    float v = row[f];
    ss += v * a_s[f];
    sd += v * a_d[f];
  }
#pragma unroll
  for (int off = 16; off; off >>= 1) {
    ss += __shfl_xor(ss, off, 32);
    sd += __shfl_xor(sd, off, 32);
  }
  if (lane == 0) { es[node] = ss; ed[node] = sd; }
}

// ---------------------------------------------------------------------------
// Pass 1: h[e] = exp(leakyrelu(es[src]+ed[dst]+b)); hsum[src] += h[e].
// ---------------------------------------------------------------------------
__global__ void gat_edge_pass1(const int* __restrict__ src,
                               const int* __restrict__ dst,
                               const float* __restrict__ es,
                               const float* __restrict__ ed,
                               const float* __restrict__ bias,
                               float* __restrict__ hbuf,
                               float* __restrict__ hsum, int E) {
  const int e = blockIdx.x * blockDim.x + threadIdx.x;
  if (e >= E) return;
  const int s = src[e], d = dst[e];
  float v = es[s] + ed[d] + bias[0];
  v = (v > 0.f) ? v : 0.05f * v;  // LeakyReLU(0.05)
  const float w = __expf(v);
  hbuf[e] = w;
  unsafeAtomicAdd(&hsum[s], w);  // global_atomic_add_f32, L2-resident
}

// ---------------------------------------------------------------------------
// Pass 2: alpha = h[e]/hsum[src]; agg[src,:] += alpha * Wx[dst,:].
// One wave per edge; 32 lanes x 8 floats = H=256. Coalesced float4 gathers,
// contiguous f32 atomics into the (L2-resident) accumulator.
// ---------------------------------------------------------------------------
__global__ void gat_edge_pass2(const int* __restrict__ src,
                               const int* __restrict__ dst,
                               const float* __restrict__ hbuf,
                               const float* __restrict__ hsum,
                               const float* __restrict__ Wx,
                               float* __restrict__ agg,
                               float* __restrict__ alphaOut, int E, int H) {
  const int lane = threadIdx.x & 31;
  const int e = blockIdx.x * (blockDim.x >> 5) + (threadIdx.x >> 5);
  if (e >= E) return;
  const int s = src[e], d = dst[e];
  const float alpha = hbuf[e] / hsum[s];
  if (alphaOut != nullptr && lane == 0) alphaOut[e] = alpha;

  const float4* wr =
      reinterpret_cast<const float4*>(Wx + (size_t)d * H) + lane * 2;
  const float4 v0 = wr[0];
  const float4 v1 = wr[1];
  float* out = agg + (size_t)s * H + lane * 8;
  unsafeAtomicAdd(out + 0, alpha * v0.x);
  unsafeAtomicAdd(out + 1, alpha * v0.y);
  unsafeAtomicAdd(out + 2, alpha * v0.z);
  unsafeAtomicAdd(out + 3, alpha * v0.w);
  unsafeAtomicAdd(out + 4, alpha * v1.x);
  unsafeAtomicAdd(out + 5, alpha * v1.y);
  unsafeAtomicAdd(out + 6, alpha * v1.z);
  unsafeAtomicAdd(out + 7, alpha * v1.w);
}

// ---------------------------------------------------------------------------
// out[i,c] = log_softmax(logits[i,:] + bfc)[c]. One wave per node, C==32.
// ---------------------------------------------------------------------------
__global__ void gat_bias_logsoftmax(const float* __restrict__ logits,
                                    const float* __restrict__ bias,
                                    float* __restrict__ out, int Nn, int C) {
  const int lane = threadIdx.x & 31;
  const int node = blockIdx.x * (blockDim.x >> 5) + (threadIdx.x >> 5);
  if (node >= Nn) return;
  float v = logits[(size_t)node * C + lane] + bias[lane];
  float m = v;
#pragma unroll
  for (int off = 16; off; off >>= 1) m = fmaxf(m, __shfl_xor(m, off, 32));
  float p = __expf(v - m);
#pragma unroll
  for (int off = 16; off; off >>= 1) p += __shfl_xor(p, off, 32);
  out[(size_t)node * C + lane] = v - m - __logf(p);
}

// ---------------------------------------------------------------------------
extern "C" void kernel_launch(void* const* d_in, const int* in_sizes, int n_in,
                              void* d_out, int out_size, void* d_ws,
                              size_t ws_size, hipStream_t stream) {
  const float* x   = (const float*)d_in[0];
  const int*   src = (const int*)d_in[1];
  const int*   dst = (const int*)d_in[2];
  const float* W1  = (const float*)d_in[3];
  const float* a1s = (const float*)d_in[4];
  const float* a1d = (const float*)d_in[5];
  const float* a1b = (const float*)d_in[6];
  const float* W2  = (const float*)d_in[7];
  const float* a2s = (const float*)d_in[8];
  const float* a2d = (const float*)d_in[9];
  const float* a2b = (const float*)d_in[10];
  const float* Wfc = (const float*)d_in[11];
  const float* bfc = (const float*)d_in[12];

  const int NF = 512, NH = 256, NC = 32;
  const int N = in_sizes[0] / NF;   // 50000
  const int E = in_sizes[1];        // 800000

  // Workspace layout (floats)
  float* wx     = (float*)d_ws;                 // [N, NH] Wx of current layer
  float* agg    = wx + (size_t)N * NH;          // [N, NH] edge aggregate
  float* esrc   = agg + (size_t)N * NH;         // [N]
  float* edst   = esrc + N;                     // [N]
  float* hsum   = edst + N;                     // [N]
  float* hbuf   = hsum + N;                     // [E]
  float* logits = hbuf + E;                     // [N, NC]

  float* lsm_out = (float*)d_out;               // [N, NC]
  float* alpha2  = lsm_out + (size_t)N * NC;    // [E]

  const int wavesPerBlk = 8;  // 256 threads

  // ---- Layer 1: Wx1 = x @ W1 ----
  {
    int tiles = (N / 16) * (NH / 64);  // CT=4 -> 16x64 per wave
    gat_gemm_f32_wmma<false, 4><<<(tiles + wavesPerBlk - 1) / wavesPerBlk, 256,
                                  0, stream>>>(x, W1, wx, N, NH, NF);
  }
  gat_gemv2<<<(N + wavesPerBlk - 1) / wavesPerBlk, 256, 0, stream>>>(
      wx, a1s, a1d, esrc, edst, N, NH);
  hipMemsetAsync(hsum, 0, (size_t)N * sizeof(float), stream);
  gat_edge_pass1<<<(E + 255) / 256, 256, 0, stream>>>(src, dst, esrc, edst,
                                                      a1b, hbuf, hsum, E);
  hipMemsetAsync(agg, 0, (size_t)N * NH * sizeof(float), stream);
  gat_edge_pass2<<<(E + wavesPerBlk - 1) / wavesPerBlk, 256, 0, stream>>>(
      src, dst, hbuf, hsum, wx, agg, nullptr, E, NH);

  // ---- Layer 2: Wx2 = relu(agg) @ W2 ----
  {
    int tiles = (N / 16) * (NH / 64);
    gat_gemm_f32_wmma<true, 4><<<(tiles + wavesPerBlk - 1) / wavesPerBlk, 256,
                                 0, stream>>>(agg, W2, wx, N, NH, NH);
  }
  gat_gemv2<<<(N + wavesPerBlk - 1) / wavesPerBlk, 256, 0, stream>>>(
      wx, a2s, a2d, esrc, edst, N, NH);
  hipMemsetAsync(hsum, 0, (size_t)N * sizeof(float), stream);
  gat_edge_pass1<<<(E + 255) / 256, 256, 0, stream>>>(src, dst, esrc, edst,
                                                      a2b, hbuf, hsum, E);
  hipMemsetAsync(agg, 0, (size_t)N * NH * sizeof(float), stream);
  gat_edge_pass2<<<(E + wavesPerBlk - 1) / wavesPerBlk, 256, 0, stream>>>(
      src, dst, hbuf, hsum, wx, agg, alpha2, E, NH);

  // ---- Classifier: logits = relu(agg) @ Wfc ; out = log_softmax(logits+b) --
  {
    int tiles = (N / 16) * (NC / 32);  // CT=2 -> 16x32 per wave
    gat_gemm_f32_wmma<true, 2><<<(tiles + wavesPerBlk - 1) / wavesPerBlk, 256,
                                 0, stream>>>(agg, Wfc, logits, N, NC, NH);
  }
  gat_bias_logsoftmax<<<(N + wavesPerBlk - 1) / wavesPerBlk, 256, 0, stream>>>(
      logits, bfc, lsm_out, N, NC);
}